// PointwiseAggregatedAttention_60172491817562
// MI455X (gfx1250) — compile-verified
//
#include <hip/hip_runtime.h>

typedef __attribute__((ext_vector_type(2))) float v2f;
typedef __attribute__((ext_vector_type(8))) float v8f;

namespace {
constexpr int Sn  = 2048;   // sequence length
constexpr int Dn  = 1024;   // model dim
constexpr int Hn  = 16;     // heads
constexpr int HDn = 64;     // head dim
constexpr int NBn = 20;     // rel-bias buckets
constexpr int KT  = 16;     // keys per tile
constexpr int QT  = 64;     // queries per workgroup
constexpr int KS_STRIDE  = 68;  // padded LDS row stride for K/V tiles (conflict-free frags)
constexpr int ATT_STRIDE = 18;  // padded (even) LDS stride for att scratch
}

// Async DMA: copy 32 contiguous bytes (2 x b128) global -> LDS, tracked by ASYNCcnt.
// ISA 15.18.3: the instruction offset is added to both the LDS and the global address.
__device__ __forceinline__ void async_copy_32B(unsigned lds_off, const float* gptr) {
    unsigned long long ga = (unsigned long long)(uintptr_t)gptr;
    asm volatile(
        "global_load_async_to_lds_b128 %0, %1, off\n\t"
        "global_load_async_to_lds_b128 %0, %1, off offset:16"
        :
        : "v"(lds_off), "v"(ga)
        : "memory");
}

__device__ __forceinline__ void wait_async0() {
    asm volatile("s_wait_asynccnt 0x0" ::: "memory");
}

__global__ __launch_bounds__(128)
void paa_fused_kernel(const float* __restrict__ v,
                      const float* __restrict__ k,
                      const float* __restrict__ q,
                      const unsigned char* __restrict__ mask,
                      const int* __restrict__ diff,
                      const float* __restrict__ rel_bias,
                      float* __restrict__ out)
{
    __shared__ float Ks[2][16 * KS_STRIDE];   // double-buffered K tile
    __shared__ float Vs[2][16 * KS_STRIDE];   // double-buffered V tile
    __shared__ float Att[4][16 * ATT_STRIDE]; // per-wave att re-fragment scratch
    __shared__ float biasL[32];

    const int b    = blockIdx.z;
    const int h    = blockIdx.y;
    const int qt   = blockIdx.x;
    const int tid  = threadIdx.x;
    const int wave = tid >> 5;
    const int lane = tid & 31;
    const int m    = lane & 15;   // M (A/C rows) or N (B/C cols) within 16
    const int half = lane >> 4;   // lane half selects K offset in A/B frags
    const int k0   = half * 2;

    // stage this head's rel_bias column (20 floats) into LDS
    if (tid < NBn) biasL[tid] = rel_bias[tid * Hn + h];

    const int qbase = qt * QT + wave * 16;
    const float scale = 0.125f; // 1/sqrt(64)

    // ---- Q strip as A-fragments for 16x16x4 f32 WMMA, pre-scaled ----
    v2f aQ[16];
    {
        const float* qp = q + ((size_t)b * Sn + qbase + m) * Dn + h * HDn + k0;
        #pragma unroll
        for (int c = 0; c < 16; ++c) {
            aQ[c].x = qp[4 * c]     * scale;
            aQ[c].y = qp[4 * c + 1] * scale;
        }
    }

    v8f o0 = {}, o1 = {}, o2 = {}, o3 = {};

    // per-lane base row of diff/mask: query rows qbase+half*8 .. +7 live in regs r=0..7
    const size_t srow0 = ((size_t)b * Sn + (size_t)(qbase + half * 8)) * (size_t)Sn;
    const int*            diff_p = diff + srow0;
    const unsigned char*  mask_p = mask + srow0;

    // cooperative K/V loader coords: 128 threads x 32B = 16x64 f32 tile
    const int lrow = tid >> 3;        // 0..15
    const int lcol = (tid & 7) * 8;   // 0,8,..,56

    // per-thread global row bases (tile kt adds kt*KT*Dn floats)
    const float* kRow0 = k + ((size_t)b * Sn + lrow) * Dn + h * HDn + lcol;
    const float* vRow0 = v + ((size_t)b * Sn + lrow) * Dn + h * HDn + lcol;

    // per-thread LDS byte addresses for both buffers (flat ptr low 32 bits = LDS offset)
    const unsigned ldsK[2] = {
        (unsigned)(uintptr_t)&Ks[0][lrow * KS_STRIDE + lcol],
        (unsigned)(uintptr_t)&Ks[1][lrow * KS_STRIDE + lcol] };
    const unsigned ldsV[2] = {
        (unsigned)(uintptr_t)&Vs[0][lrow * KS_STRIDE + lcol],
        (unsigned)(uintptr_t)&Vs[1][lrow * KS_STRIDE + lcol] };

    constexpr int NT = Sn / KT; // 128 key tiles

    // prologue: async-stage tile 0 into buffer 0
    async_copy_32B(ldsK[0], kRow0);
    async_copy_32B(ldsV[0], vRow0);

    for (int kt = 0; kt < NT; ++kt) {
        const int cur = kt & 1;
        const int kbase = kt * KT;

        wait_async0();     // my async loads for buf[cur] landed
        __syncthreads();   // everyone's landed; everyone done reading buf[cur^1]

        // issue async stage of the next tile into the other buffer (overlaps compute)
        if (kt + 1 < NT) {
            const size_t adv = (size_t)(kt + 1) * KT * Dn;
            async_copy_32B(ldsK[cur ^ 1], kRow0 + adv);
            async_copy_32B(ldsV[cur ^ 1], vRow0 + adv);
        }

        const float* KsC = &Ks[cur][0];
        const float* VsC = &Vs[cur][0];

        // ---- scores: S16x16 = Qstrip(16x64) @ K_tile^T(64x16), K-dim chained ----
        v8f s = {};
        #pragma unroll
        for (int c = 0; c < 16; ++c) {
            // B frag: B[K=4c+k0+{0,1}][N=m] == Ks[key=m][hd=4c+k0+{0,1}] -> contiguous b64
            v2f bK = *(const v2f*)&KsC[m * KS_STRIDE + 4 * c + k0];
            s = __builtin_amdgcn_wmma_f32_16x16x4_f32(false, aQ[c], false, bK,
                                                      (short)0, s, false, false);
        }

        // ---- elementwise: bias gather + silu + mask; spill att to LDS scratch ----
        #pragma unroll
        for (int r = 0; r < 8; ++r) {
            const size_t off = (size_t)r * Sn + (size_t)(kbase + m);
            const int   dix = diff_p[off];
            const float bv  = biasL[dix];
            const float x   = s[r] + bv;
            const float sg  = __builtin_amdgcn_rcpf(1.0f + __expf(-x));
            const float att = x * sg * (float)mask_p[off];
            Att[wave][(r + half * 8) * ATT_STRIDE + m] = att;
        }

        // ---- re-fragment att (C-layout -> A-layout) from own wave's scratch ----
        v2f aA[4];
        #pragma unroll
        for (int c2 = 0; c2 < 4; ++c2) {
            aA[c2] = *(const v2f*)&Att[wave][m * ATT_STRIDE + 4 * c2 + k0];
        }

        // ---- O(16x64) += att(16x16) @ V_tile(16x64), 4 N-tiles x 4 K-chunks ----
        #pragma unroll
        for (int c2 = 0; c2 < 4; ++c2) {
            const int kr = 4 * c2 + k0;
            v2f b0, b1, b2, b3;
            b0.x = VsC[kr * KS_STRIDE +  0 + m]; b0.y = VsC[(kr + 1) * KS_STRIDE +  0 + m];
            b1.x = VsC[kr * KS_STRIDE + 16 + m]; b1.y = VsC[(kr + 1) * KS_STRIDE + 16 + m];
            b2.x = VsC[kr * KS_STRIDE + 32 + m]; b2.y = VsC[(kr + 1) * KS_STRIDE + 32 + m];
            b3.x = VsC[kr * KS_STRIDE + 48 + m]; b3.y = VsC[(kr + 1) * KS_STRIDE + 48 + m];
            o0 = __builtin_amdgcn_wmma_f32_16x16x4_f32(false, aA[c2], false, b0, (short)0, o0, false, false);
            o1 = __builtin_amdgcn_wmma_f32_16x16x4_f32(false, aA[c2], false, b1, (short)0, o1, false, false);
            o2 = __builtin_amdgcn_wmma_f32_16x16x4_f32(false, aA[c2], false, b2, (short)0, o2, false, false);
            o3 = __builtin_amdgcn_wmma_f32_16x16x4_f32(false, aA[c2], false, b3, (short)0, o3, false, false);
        }
        // no trailing barrier: next iteration's wait+barrier provides the epoch
    }

    // ---- store O: out[b, qrow, h*64 + d] ----
    #pragma unroll
    for (int r = 0; r < 8; ++r) {
        float* op = out + ((size_t)b * Sn + qbase + half * 8 + r) * Dn + h * HDn + m;
        op[0]  = o0[r];
        op[16] = o1[r];
        op[32] = o2[r];
        op[48] = o3[r];
    }
}

extern "C" void kernel_launch(void* const* d_in, const int* in_sizes, int n_in,
                              void* d_out, int out_size, void* d_ws, size_t ws_size,
                              hipStream_t stream) {
    (void)in_sizes; (void)n_in; (void)out_size; (void)d_ws; (void)ws_size;
    const float*         v        = (const float*)d_in[0];
    const float*         k        = (const float*)d_in[1];
    const float*         q        = (const float*)d_in[2];
    const unsigned char* mask     = (const unsigned char*)d_in[3];  // jnp.bool_ -> 1 byte
    const int*           diff     = (const int*)d_in[4];
    const float*         rel_bias = (const float*)d_in[5];
    float*               out      = (float*)d_out;

    dim3 grid(Sn / QT, Hn, 2 /*B*/);
    dim3 block(128);
    paa_fused_kernel<<<grid, block, 0, stream>>>(v, k, q, mask, diff, rel_bias, out);
}